// DFMAtt_10969346474548
// MI455X (gfx1250) — compile-verified
//
#include <hip/hip_runtime.h>
#include <hip/hip_bf16.h>

#define BATCH 8
#define CIN   256
#define OCH   256
#define HH    96
#define WW    96
#define HWSZ  (HH * WW)          // 9216
#define NPIX  (BATCH * HWSZ)     // 73728
#define KF    4

typedef __attribute__((ext_vector_type(2))) float v2f;
typedef __attribute__((ext_vector_type(8))) float v8f;

// ---------------------------------------------------------------------------
// Kernel 1: pack flow/logit weights into a zero-padded 16x256 block + 16 biases
//   rows 0..7  : Woff (k*2+axis, c)   -> flow projections
//   rows 8..11 : Wwt  (k, c)          -> logit projections
//   rows 12..15: zero
// ---------------------------------------------------------------------------
__global__ __launch_bounds__(256)
void pack_aux_weights(const float* __restrict__ Woff, const float* __restrict__ boff,
                      const float* __restrict__ Wwt,  const float* __restrict__ bwt,
                      float* __restrict__ Waux, float* __restrict__ baux)
{
    const int i = blockIdx.x * blockDim.x + threadIdx.x;
    if (i < 2048) {                       // rows 0..7 (K*2 x C flattened)
        Waux[i] = Woff[i];
    } else if (i < 3072) {                // rows 8..11
        Waux[i] = Wwt[i - 2048];
    } else if (i < 4096) {                // rows 12..15 zero pad
        Waux[i] = 0.0f;
    } else if (i < 4096 + 16) {
        const int j = i - 4096;
        float v = 0.0f;
        if (j < 8)       v = boff[j];
        else if (j < 12) v = bwt[j - 8];
        baux[j] = v;
    }
}

// ---------------------------------------------------------------------------
// Kernel 2: fused per-pixel GEMM using V_WMMA_F32_16X16X4_F32 (wave32).
//   D(16 pix x 16 outch) += A(16 pix x 4 ch) * B(4 ch x 16 outch)
// A layout (ISA 7.12.2, 32-bit A 16x4): lanes 0-15 -> K=0/1 in v0/v1,
//                                       lanes16-31 -> K=2/3.
// B layout mirrors (lane&15 = N, K halves split across lane halves).
// C/D layout: VGPR v holds row M = v + 8*(lane>>4), col N = lane&15.
// grid = (NPIX/16, 17): y<16 -> proj tile (o0=16y), y==16 -> aux tile.
// proj written pixel-major [NPIX][256] for coalesced gathers later.
// ---------------------------------------------------------------------------
__global__ __launch_bounds__(32)
void fused_proj_gemm(const float* __restrict__ x,     // [B][CIN][HW]
                     const float* __restrict__ Wc,    // [OCH][CIN]
                     const float* __restrict__ bc,    // [OCH]
                     const float* __restrict__ Waux,  // [16][CIN]
                     const float* __restrict__ baux,  // [16]
                     float* __restrict__ proj,        // [NPIX][OCH]
                     float* __restrict__ aux)         // [NPIX][16]
{
    const int lane = threadIdx.x;
    const int n16  = lane & 15;
    const int half = lane >> 4;

    const bool  isAux = (blockIdx.y == 16);
    const float* Wm   = isAux ? Waux : Wc;
    const float* bias = isAux ? baux : bc;
    float*       dst  = isAux ? aux  : proj;
    const int    OC   = isAux ? 16   : OCH;
    const int    o0   = isAux ? 0    : (int)blockIdx.y * 16;

    const int p0  = (int)blockIdx.x * 16;       // global pixel index
    const int b   = p0 / HWSZ;                  // tiles never straddle batches
    const int pp0 = p0 - b * HWSZ;
    const float* xb = x + (size_t)b * CIN * HWSZ + pp0 + n16;  // lane's pixel column

    const float* wrow = Wm + (size_t)(o0 + n16) * CIN;         // lane's outch row

    v8f acc = {};
    #pragma unroll 4
    for (int c0 = 0; c0 < CIN; c0 += 4) {
        const int ka = c0 + half * 2;           // this lane-half's K pair
        v2f a;
        a.x = xb[(size_t)ka * HWSZ];            // A[m][ka]
        a.y = xb[(size_t)(ka + 1) * HWSZ];      // A[m][ka+1]
        v2f bm = *(const v2f*)(wrow + ka);      // B[ka][n], B[ka+1][n] (8B aligned)
        acc = __builtin_amdgcn_wmma_f32_16x16x4_f32(
                  false, a, false, bm, (short)0, acc, false, false);
    }

    const float bo = bias[o0 + n16];
    float* op = dst + (size_t)p0 * OC + o0 + n16;
    #pragma unroll
    for (int v = 0; v < 8; ++v) {
        const int m = v + half * 8;             // pixel row of this VGPR
        op[(size_t)m * OC] = acc[v] + bo;
    }
}

// ---------------------------------------------------------------------------
// Kernel 3: softmax over K logits, K bilinear taps (zeros padding,
// align_corners=False) on pixel-major proj, weighted sum -> out [B][OCH][HW].
// One block per pixel, thread = output channel -> proj reads fully coalesced
// (256 contiguous floats per tap), and proj (75.5 MB) is L2-resident.
// ---------------------------------------------------------------------------
__global__ __launch_bounds__(256)
void sample_blend(const float* __restrict__ proj,  // [NPIX][OCH]
                  const float* __restrict__ aux,   // [NPIX][16]
                  float* __restrict__ out)         // [B][OCH][HW]
{
    const int p  = blockIdx.x;
    const int t  = threadIdx.x;                  // output channel
    const int b  = p / HWSZ;
    const int pp = p - b * HWSZ;
    const int yy = pp / WW;
    const int xx = pp - yy * WW;

    const float* a = aux + (size_t)p * 16;

    // softmax over logits a[8..11]
    const float l0 = a[8], l1 = a[9], l2 = a[10], l3 = a[11];
    const float mx = fmaxf(fmaxf(l0, l1), fmaxf(l2, l3));
    const float e0 = expf(l0 - mx), e1 = expf(l1 - mx);
    const float e2 = expf(l2 - mx), e3 = expf(l3 - mx);
    const float inv = 1.0f / (e0 + e1 + e2 + e3);
    const float wk[KF] = { e0 * inv, e1 * inv, e2 * inv, e3 * inv };

    // ix = (x + flow_x) * W/(W-1) - 0.5  (align_corners=False unnormalize)
    const float sxf = (float)WW / (float)(WW - 1);
    const float syf = (float)HH / (float)(HH - 1);

    const float* pb = proj + (size_t)b * HWSZ * OCH + t;

    auto tap = [&](int xi, int yi, float wgt) -> float {
        const bool v  = (xi >= 0) && (xi < WW) && (yi >= 0) && (yi < HH);
        const int  xc = min(max(xi, 0), WW - 1);
        const int  yc = min(max(yi, 0), HH - 1);
        const float g = pb[(size_t)(yc * WW + xc) * OCH];
        return v ? g * wgt : 0.0f;
    };

    float acc = 0.0f;
    #pragma unroll
    for (int k = 0; k < KF; ++k) {
        const float ix = ((float)xx + a[2 * k + 0]) * sxf - 0.5f;
        const float iy = ((float)yy + a[2 * k + 1]) * syf - 0.5f;
        const float x0f = floorf(ix), y0f = floorf(iy);
        const float wx1 = ix - x0f,  wx0 = 1.0f - wx1;
        const float wy1 = iy - y0f,  wy0 = 1.0f - wy1;
        const int x0 = (int)x0f, y0 = (int)y0f;
        const int x1 = x0 + 1,   y1 = y0 + 1;

        float s = tap(x0, y0, wx0 * wy0) + tap(x1, y0, wx1 * wy0)
                + tap(x0, y1, wx0 * wy1) + tap(x1, y1, wx1 * wy1);
        acc += wk[k] * s;
    }

    out[(size_t)b * OCH * HWSZ + (size_t)t * HWSZ + pp] = acc;
}

// ---------------------------------------------------------------------------
extern "C" void kernel_launch(void* const* d_in, const int* in_sizes, int n_in,
                              void* d_out, int out_size, void* d_ws, size_t ws_size,
                              hipStream_t stream)
{
    const float* x    = (const float*)d_in[0];   // [8,256,96,96]
    const float* Wc   = (const float*)d_in[1];   // [256,256]
    const float* bc   = (const float*)d_in[2];   // [256]
    const float* Woff = (const float*)d_in[3];   // [4,2,256]
    const float* boff = (const float*)d_in[4];   // [4,2]
    const float* Wwt  = (const float*)d_in[5];   // [4,256]
    const float* bwt  = (const float*)d_in[6];   // [4]
    float* out = (float*)d_out;                  // [8,256,96,96]

    float* ws   = (float*)d_ws;
    float* proj = ws;                                   // NPIX*256 floats (75.5 MB)
    float* aux  = proj + (size_t)NPIX * OCH;            // NPIX*16  floats (4.7 MB)
    float* Waux = aux  + (size_t)NPIX * 16;             // 4096 floats
    float* baux = Waux + 4096;                          // 16 floats

    pack_aux_weights<<<17, 256, 0, stream>>>(Woff, boff, Wwt, bwt, Waux, baux);
    fused_proj_gemm<<<dim3(NPIX / 16, 17), 32, 0, stream>>>(
        x, Wc, bc, Waux, baux, proj, aux);
    sample_blend<<<NPIX, 256, 0, stream>>>(proj, aux, out);
}